// GAT_73512660238647
// MI455X (gfx1250) — compile-verified
//
#include <hip/hip_runtime.h>
#include <math.h>

#define DIN   128
#define DH    64
#define HEADS 2
#define DOUT  64
#define NEG_SLOPE 0.2f

typedef float v2f __attribute__((ext_vector_type(2)));
typedef float v8f __attribute__((ext_vector_type(8)));

__device__ __forceinline__ void atomic_max_f32(float* addr, float v) {
  if (v >= 0.0f) atomicMax((int*)addr, __float_as_int(v));
  else           atomicMin((unsigned int*)addr, __float_as_uint(v));
}

// ---------------------------------------------------------------- fills
__global__ void fill_f32(float* __restrict__ p, long n, float v) {
  long i = (long)blockIdx.x * blockDim.x + threadIdx.x;
  if (i < n) p[i] = v;
}

// ---------------------------------------------------------------- GEMM (fp32 WMMA)
// C[M,Nc] = A[M,K] @ B[K,Nc], one wave per 16x16 tile, V_WMMA_F32_16X16X4_F32.
__global__ void __launch_bounds__(256)
gemm_wmma_f32(const float* __restrict__ A, const float* __restrict__ B,
              float* __restrict__ C, int M, int K, int Nc) {
  int gid  = blockIdx.x * blockDim.x + threadIdx.x;
  int wave = gid >> 5;
  int lane = threadIdx.x & 31;
  int tilesN = Nc >> 4;
  int tm = wave / tilesN;
  int tn = wave - tm * tilesN;
  if (tm * 16 >= M) return;            // wave-uniform: EXEC stays all-ones
  int half = lane >> 4;                // selects K pair {0,1} vs {2,3}
  int l16  = lane & 15;
  long rowA = (long)tm * 16 + l16;     // A: M across lanes 0..15
  if (rowA >= M) rowA = M - 1;         // clamp (M is a multiple of 16 here)
  int  colB = tn * 16 + l16;           // B/C: N across lanes 0..15
  v8f acc = {};
  for (int k0 = 0; k0 < K; k0 += 4) {
    int ka = k0 + half * 2;
    v2f a, b;
    a.x = A[rowA * K + ka];
    a.y = A[rowA * K + ka + 1];
    b.x = B[(long)ka       * Nc + colB];
    b.y = B[(long)(ka + 1) * Nc + colB];
    acc = __builtin_amdgcn_wmma_f32_16x16x4_f32(false, a, false, b,
                                                (short)0, acc, false, false);
  }
#pragma unroll
  for (int r = 0; r < 8; ++r) {
    int row = tm * 16 + r + half * 8;  // C/D layout: lanes>=16 hold M=8..15
    if (row < M) C[(long)row * Nc + colB] = acc[r];
  }
}

// ---------------------------------------------------------------- attention dots
// a_src[n,h] = sum_c h[n,h,c]*att_src[h,c] ; same for a_dst.
__global__ void att_dots(const float* __restrict__ h,
                         const float* __restrict__ att_s,
                         const float* __restrict__ att_d,
                         float* __restrict__ as_, float* __restrict__ ad_,
                         int n, int heads, int C) {
  int i = blockIdx.x * blockDim.x + threadIdx.x;
  if (i >= n * heads) return;
  int node = i / heads, hd = i - node * heads;
  const float* hp = h + ((long)node * heads + hd) * C;
  const float* sp = att_s + hd * C;
  const float* dp = att_d + hd * C;
  float ss = 0.f, dd = 0.f;
  for (int c = 0; c < C; ++c) { float v = hp[c]; ss += v * sp[c]; dd += v * dp[c]; }
  as_[i] = ss; ad_[i] = dd;
}

// ---------------------------------------------------------------- edge pass 1: logits + segment max
__global__ void edge_max(const int* __restrict__ srcs, const int* __restrict__ dsts,
                         int E0, int Etot,
                         const float* __restrict__ as_, const float* __restrict__ ad_,
                         float* __restrict__ ev, float* __restrict__ m, int heads) {
  int e = blockIdx.x * blockDim.x + threadIdx.x;
  if (e >= Etot) return;
  int s = (e < E0) ? srcs[e] : (e - E0);
  int d = (e < E0) ? dsts[e] : (e - E0);
  for (int h = 0; h < heads; ++h) {
    float v = as_[s * heads + h] + ad_[d * heads + h];
    v = (v > 0.f) ? v : NEG_SLOPE * v;                  // leaky_relu
    ev[(long)e * heads + h] = v;
    atomic_max_f32(&m[d * heads + h], v);
  }
}

// ---------------------------------------------------------------- edge pass 2: exp + segment sum
__global__ void edge_sum(const int* __restrict__ dsts, int E0, int Etot,
                         float* __restrict__ ev, const float* __restrict__ m,
                         float* __restrict__ den, int heads) {
  int e = blockIdx.x * blockDim.x + threadIdx.x;
  if (e >= Etot) return;
  int d = (e < E0) ? dsts[e] : (e - E0);
  for (int h = 0; h < heads; ++h) {
    float ex = expf(ev[(long)e * heads + h] - m[d * heads + h]);
    ev[(long)e * heads + h] = ex;
    atomicAdd(&den[d * heads + h], ex);
  }
}

// ---------------------------------------------------------------- edge pass 3: weighted scatter-add
// one wave per edge; lanes stride the feature dim.
__global__ void edge_agg(const int* __restrict__ srcs, const int* __restrict__ dsts,
                         int E0, int Etot,
                         const float* __restrict__ ev, const float* __restrict__ den,
                         const float* __restrict__ h, float* __restrict__ out,
                         int heads, int C) {
  long gid = (long)blockIdx.x * blockDim.x + threadIdx.x;
  int e = (int)(gid >> 5);
  int lane = (int)(gid & 31);
  if (e >= Etot) return;
  int s = (e < E0) ? srcs[e] : (e - E0);
  int d = (e < E0) ? dsts[e] : (e - E0);
  int F = heads * C;
  for (int c = lane; c < F; c += 32) {
    int hd = c / C;
    float alpha = ev[(long)e * heads + hd] / den[d * heads + hd];
    atomicAdd(&out[(long)d * F + c], h[(long)s * F + c] * alpha);
  }
}

// ---------------------------------------------------------------- epilogues
__global__ void elu_bias(float* __restrict__ p, const float* __restrict__ bias,
                         long n, int F) {
  long i = (long)blockIdx.x * blockDim.x + threadIdx.x;
  if (i >= n) return;
  float v = p[i] + bias[i % F];
  p[i] = (v > 0.f) ? v : (expf(v) - 1.f);               // ELU(alpha=1)
}

__global__ void bias_add(float* __restrict__ p, const float* __restrict__ bias,
                         long n, int F) {
  long i = (long)blockIdx.x * blockDim.x + threadIdx.x;
  if (i >= n) return;
  p[i] += bias[i % F];
}

// ================================================================ host
extern "C" void kernel_launch(void* const* d_in, const int* in_sizes, int n_in,
                              void* d_out, int out_size, void* d_ws, size_t ws_size,
                              hipStream_t stream) {
  const float* x    = (const float*)d_in[0];
  const int*   ei   = (const int*)  d_in[1];
  const float* W1   = (const float*)d_in[2];
  const float* atS1 = (const float*)d_in[3];
  const float* atD1 = (const float*)d_in[4];
  const float* b1   = (const float*)d_in[5];
  const float* W2   = (const float*)d_in[6];
  const float* atS2 = (const float*)d_in[7];
  const float* atD2 = (const float*)d_in[8];
  const float* b2   = (const float*)d_in[9];
  float* out = (float*)d_out;

  const int N    = in_sizes[0] / DIN;
  const int E0   = in_sizes[1] / 2;
  const int Etot = E0 + N;
  const int* srcs = ei;
  const int* dsts = ei + E0;

  // ---- workspace carve-up (floats)
  float* ws = (float*)d_ws;
  long off = 0;
  float* h1  = ws + off; off += (long)N * (HEADS * DH);   // [N,128]
  float* o1  = ws + off; off += (long)N * (HEADS * DH);   // [N,128] aggregated
  float* ex1 = ws + off; off += (long)Etot * HEADS;       // per-edge logits/exp
  float* a1s = ws + off; off += (long)N * HEADS;
  float* a1d = ws + off; off += (long)N * HEADS;
  float* m1  = ws + off; off += (long)N * HEADS;
  float* d1  = ws + off; off += (long)N * HEADS;
  // layer-2 buffers alias h1 (dead after layer-1 aggregation)
  float* h2  = h1;                       // [N,64]
  float* ex2 = h1 + (long)N * DOUT;      // [Etot]
  float* a2s = ex2 + Etot;
  float* a2d = a2s + N;
  float* m2  = a2d + N;
  float* d2  = m2 + N;

  const int TB = 256;
  auto blocks = [](long n, int tb) { return (unsigned)((n + tb - 1) / tb); };

  // ======== layer 1 ========
  fill_f32<<<blocks((long)N * HEADS * DH, TB), TB, 0, stream>>>(o1, (long)N * HEADS * DH, 0.f);
  fill_f32<<<blocks((long)N * HEADS, TB), TB, 0, stream>>>(d1, (long)N * HEADS, 0.f);
  fill_f32<<<blocks((long)N * HEADS, TB), TB, 0, stream>>>(m1, (long)N * HEADS, -INFINITY);

  {
    long waves = (long)((N + 15) / 16) * ((HEADS * DH) / 16);
    gemm_wmma_f32<<<blocks(waves * 32, TB), TB, 0, stream>>>(x, W1, h1, N, DIN, HEADS * DH);
  }
  att_dots<<<blocks((long)N * HEADS, TB), TB, 0, stream>>>(h1, atS1, atD1, a1s, a1d, N, HEADS, DH);
  edge_max<<<blocks(Etot, TB), TB, 0, stream>>>(srcs, dsts, E0, Etot, a1s, a1d, ex1, m1, HEADS);
  edge_sum<<<blocks(Etot, TB), TB, 0, stream>>>(dsts, E0, Etot, ex1, m1, d1, HEADS);
  edge_agg<<<blocks((long)Etot * 32, TB), TB, 0, stream>>>(srcs, dsts, E0, Etot, ex1, d1, h1, o1, HEADS, DH);
  elu_bias<<<blocks((long)N * HEADS * DH, TB), TB, 0, stream>>>(o1, b1, (long)N * HEADS * DH, HEADS * DH);

  // ======== layer 2 ========
  fill_f32<<<blocks((long)N * DOUT, TB), TB, 0, stream>>>(out, (long)N * DOUT, 0.f);
  fill_f32<<<blocks(N, TB), TB, 0, stream>>>(d2, N, 0.f);
  fill_f32<<<blocks(N, TB), TB, 0, stream>>>(m2, N, -INFINITY);

  {
    long waves = (long)((N + 15) / 16) * (DOUT / 16);
    gemm_wmma_f32<<<blocks(waves * 32, TB), TB, 0, stream>>>(o1, W2, h2, N, HEADS * DH, DOUT);
  }
  att_dots<<<blocks(N, TB), TB, 0, stream>>>(h2, atS2, atD2, a2s, a2d, N, 1, DOUT);
  edge_max<<<blocks(Etot, TB), TB, 0, stream>>>(srcs, dsts, E0, Etot, a2s, a2d, ex2, m2, 1);
  edge_sum<<<blocks(Etot, TB), TB, 0, stream>>>(dsts, E0, Etot, ex2, m2, d2, 1);
  edge_agg<<<blocks((long)Etot * 32, TB), TB, 0, stream>>>(srcs, dsts, E0, Etot, ex2, d2, h2, out, 1, DOUT);
  bias_add<<<blocks((long)N * DOUT, TB), TB, 0, stream>>>(out, b2, (long)N * DOUT, DOUT);
}